// GraphConvolution_4252017623097
// MI455X (gfx1250) — compile-verified
//
#include <hip/hip_runtime.h>
#include <hip/hip_bf16.h>

typedef __attribute__((ext_vector_type(2))) float v2f;
typedef __attribute__((ext_vector_type(8))) float v8f;
typedef __attribute__((ext_vector_type(4))) int   v4i;

typedef __attribute__((address_space(1))) v4i* gas_v4i;   // global (device) AS
typedef __attribute__((address_space(3))) v4i* las_v4i;   // LDS AS

#define N_NODES 50000
#define N_EDGES 800000
#define D_IN    512
#define D_OUT   128

// Feature probes (keep source compiling on any toolchain revision).
#if defined(__has_builtin)
#  if __has_builtin(__builtin_amdgcn_global_load_async_to_lds_b128)
#    define HAVE_ASYNC_LDS 1
#  endif
#  if __has_builtin(__builtin_amdgcn_s_wait_asynccnt)
#    define HAVE_WAIT_ASYNC 1
#  endif
#endif
#ifndef HAVE_ASYNC_LDS
#  define HAVE_ASYNC_LDS 0
#endif
#ifndef HAVE_WAIT_ASYNC
#  define HAVE_WAIT_ASYNC 0
#endif

__device__ __forceinline__ void wait_asynccnt0() {
#if HAVE_WAIT_ASYNC
  __builtin_amdgcn_s_wait_asynccnt(0);
#else
  asm volatile("s_wait_asynccnt 0" ::: "memory");
#endif
}

// ---------------------------------------------------------------------------
// Pre-pass: pack W into k-pair-interleaved layout so a WMMA B fragment
// {W[k][n], W[k+1][n]} is one contiguous 8B load:
//   wp[(k/2)*D_OUT + n] = { W[k][n], W[k+1][n] }          (256 KB total)
// ---------------------------------------------------------------------------
__global__ void gcn_wpack(const float* __restrict__ w, float2* __restrict__ wp) {
  const int i = blockIdx.x * blockDim.x + threadIdx.x;   // pair index
  if (i < (D_IN / 2) * D_OUT) {
    const int p = i / D_OUT;
    const int n = i - p * D_OUT;
    wp[i] = make_float2(w[(size_t)(2 * p) * D_OUT + n],
                        w[(size_t)(2 * p + 1) * D_OUT + n]);
  }
}

// ---------------------------------------------------------------------------
// Kernel 1: support = X @ W using V_WMMA_F32_16X16X4_F32 (exact fp32 path).
// Block = 256 threads = 8 waves. Each wave computes a 32(M) x 128(N) strip:
// two 16-row WMMA tiles sharing every B fragment. Packed W staged through
// LDS in 64KB tiles via the gfx1250 async global->LDS path; each B fragment
// is a single aligned ds_load_b64 (no re-pairing movs).
// ---------------------------------------------------------------------------
__global__ __launch_bounds__(256) void gcn_gemm_wmma(
    const float* __restrict__ x, const float2* __restrict__ wp,
    float* __restrict__ support)
{
  __shared__ __align__(16) float2 w_tile[64 * D_OUT];   // 64 KB, pair rows

  const int tid  = threadIdx.x;
  const int wave = tid >> 5;
  const int lane = tid & 31;
  const int ln   = lane & 15;
  const int hi   = lane >> 4;             // 0: lanes 0-15, 1: lanes 16-31

  const int m0 = blockIdx.x * 256 + wave * 32;
  // Clamp rows for A loads: WMMA needs EXEC all-1s; duplicated rows are
  // harmless because out-of-range stores are guarded below.
  const int r0 = (m0 + ln      < N_NODES) ? (m0 + ln)      : (N_NODES - 1);
  const int r1 = (m0 + 16 + ln < N_NODES) ? (m0 + 16 + ln) : (N_NODES - 1);
  const float* xrow0 = x + (size_t)r0 * D_IN;
  const float* xrow1 = x + (size_t)r1 * D_IN;

  v8f acc0[8], acc1[8];
  #pragma unroll
  for (int t = 0; t < 8; ++t) { acc0[t] = (v8f)0.0f; acc1[t] = (v8f)0.0f; }

  for (int kb = 0; kb < D_IN; kb += 128) {
    // Stage packed W rows [kb/2, kb/2+64) -> LDS (4096 x 16B / 256 threads).
    // Pair layout is preserved verbatim by the async B128 DMA.
    const float4* wsrc = (const float4*)(wp + (size_t)(kb / 2) * D_OUT);
    float4* wdst = (float4*)w_tile;
#if HAVE_ASYNC_LDS
    #pragma unroll
    for (int i = 0; i < 16; ++i) {
      __builtin_amdgcn_global_load_async_to_lds_b128(
          (gas_v4i)(wsrc + tid + i * 256),
          (las_v4i)(wdst + tid + i * 256),
          0, 0);
    }
    wait_asynccnt0();
#else
    #pragma unroll
    for (int i = 0; i < 16; ++i)
      wdst[tid + i * 256] = wsrc[tid + i * 256];
#endif
    __syncthreads();

    for (int kk = 0; kk < 128; kk += 4) {
      // A fragments 16x4 f32: lane(<16) holds {K=kk,kk+1}, lane(>=16) {kk+2,kk+3}
      v2f a0 = *(const v2f*)(xrow0 + kb + kk + 2 * hi);
      v2f a1 = *(const v2f*)(xrow1 + kb + kk + 2 * hi);
      const int prow = (kk >> 1) + hi;    // (kk + 2*hi)/2
      #pragma unroll
      for (int t = 0; t < 8; ++t) {
        // B fragment 4x16 f32: one contiguous 8B LDS load per fragment,
        // shared by both M-tiles -> 2 WMMAs per ds_load_b64.
        v2f b = *(const v2f*)&w_tile[prow * D_OUT + t * 16 + ln];
        acc0[t] = __builtin_amdgcn_wmma_f32_16x16x4_f32(
            false, a0, false, b, (short)0, acc0[t], false, false);
        acc1[t] = __builtin_amdgcn_wmma_f32_16x16x4_f32(
            false, a1, false, b, (short)0, acc1[t], false, false);
      }
    }
    __syncthreads();
  }

  // C/D layout: VGPR r -> row base + r (lanes 0-15) / base + 8 + r (16-31),
  // column = t*16 + ln.
  #pragma unroll
  for (int r = 0; r < 8; ++r) {
    const int row_a = m0 + r + 8 * hi;
    if (row_a < N_NODES) {
      #pragma unroll
      for (int t = 0; t < 8; ++t)
        support[(size_t)row_a * D_OUT + t * 16 + ln] = acc0[t][r];
    }
    const int row_b = m0 + 16 + r + 8 * hi;
    if (row_b < N_NODES) {
      #pragma unroll
      for (int t = 0; t < 8; ++t)
        support[(size_t)row_b * D_OUT + t * 16 + ln] = acc1[t][r];
    }
  }
}

// ---------------------------------------------------------------------------
// Kernel 2: out = A @ support (gather + weighted segment-sum, dst sorted).
// One wave per 256 contiguous edges; each lane owns 4 columns (32*4 = 128).
// Register accumulation while dst unchanged; atomic flush on segment change.
// ---------------------------------------------------------------------------
#define EDGES_PER_WAVE 256

__device__ __forceinline__ void flush_seg(float* __restrict__ out, int dstrow,
                                          int lane, float4 acc) {
  float* o = out + (size_t)dstrow * D_OUT + lane * 4;
  __hip_atomic_fetch_add(o + 0, acc.x, __ATOMIC_RELAXED, __HIP_MEMORY_SCOPE_AGENT);
  __hip_atomic_fetch_add(o + 1, acc.y, __ATOMIC_RELAXED, __HIP_MEMORY_SCOPE_AGENT);
  __hip_atomic_fetch_add(o + 2, acc.z, __ATOMIC_RELAXED, __HIP_MEMORY_SCOPE_AGENT);
  __hip_atomic_fetch_add(o + 3, acc.w, __ATOMIC_RELAXED, __HIP_MEMORY_SCOPE_AGENT);
}

__global__ __launch_bounds__(256) void gcn_spmm(
    const float* __restrict__ support, const int* __restrict__ esrc,
    const int* __restrict__ edst, const float* __restrict__ eval,
    float* __restrict__ out)
{
  const int lane  = threadIdx.x & 31;
  const int gwave = (blockIdx.x * blockDim.x + threadIdx.x) >> 5;
  long e0 = (long)gwave * EDGES_PER_WAVE;
  if (e0 >= N_EDGES) return;
  long e1 = e0 + EDGES_PER_WAVE;
  if (e1 > N_EDGES) e1 = N_EDGES;

  float4 acc = make_float4(0.f, 0.f, 0.f, 0.f);
  int cur = edst[e0];                       // wave-uniform
  for (long e = e0; e < e1; ++e) {
    const int d = edst[e];
    if (d != cur) {                         // uniform branch (same e all lanes)
      flush_seg(out, cur, lane, acc);
      acc = make_float4(0.f, 0.f, 0.f, 0.f);
      cur = d;
    }
    const int   s = esrc[e];
    const float v = eval[e];
    // 32 lanes x 16B = one coalesced 512B row read; support is L2-resident.
    const float4 row = *(const float4*)(support + (size_t)s * D_OUT + lane * 4);
    acc.x += row.x * v;
    acc.y += row.y * v;
    acc.z += row.z * v;
    acc.w += row.w * v;
  }
  flush_seg(out, cur, lane, acc);
}

// ---------------------------------------------------------------------------
// Zero the output (harness poisons d_out; atomics need a clean base).
// ---------------------------------------------------------------------------
__global__ void gcn_zero(float4* __restrict__ out, int n4) {
  int i = blockIdx.x * blockDim.x + threadIdx.x;
  const int stride = gridDim.x * blockDim.x;
  for (; i < n4; i += stride) out[i] = make_float4(0.f, 0.f, 0.f, 0.f);
}

extern "C" void kernel_launch(void* const* d_in, const int* in_sizes, int n_in,
                              void* d_out, int out_size, void* d_ws, size_t ws_size,
                              hipStream_t stream) {
  const float* x    = (const float*)d_in[0];
  const int*   esrc = (const int*)d_in[1];
  const int*   edst = (const int*)d_in[2];
  const float* eval = (const float*)d_in[3];
  const float* w    = (const float*)d_in[4];
  float* out = (float*)d_out;

  // Workspace layout: [0, 256KB) packed W ; [256KB, 256KB+25.6MB) support
  float2* wp      = (float2*)d_ws;
  float*  support = (float*)((char*)d_ws + (size_t)(D_IN / 2) * D_OUT * sizeof(float2));

  // 0) pack W into pair-interleaved layout (32768 pairs)
  const int npairs = (D_IN / 2) * D_OUT;
  gcn_wpack<<<(npairs + 255) / 256, 256, 0, stream>>>(w, wp);

  // 1) support = X @ W   (196 blocks x 256 rows)
  dim3 gemm_grid((N_NODES + 255) / 256);
  gcn_gemm_wmma<<<gemm_grid, 256, 0, stream>>>(x, wp, support);

  // 2) zero output
  const int n4 = (N_NODES * D_OUT) / 4;
  gcn_zero<<<1024, 256, 0, stream>>>((float4*)out, n4);

  // 3) out = segment_sum(support[src] * val, dst)
  const int nwaves = (N_EDGES + EDGES_PER_WAVE - 1) / EDGES_PER_WAVE;
  dim3 spmm_grid((nwaves + 7) / 8);
  gcn_spmm<<<spmm_grid, 256, 0, stream>>>(support, esrc, edst, eval, out);
}